// MinimalA2AAttnOp_10668698763931
// MI455X (gfx1250) — compile-verified
//
#include <hip/hip_runtime.h>
#include <math.h>

// Problem constants (from reference: B=2, S=4096, H=16, D=64, BLOCK=64, topk=ceil(0.1*64)=7)
#define BB   2
#define HH   16
#define SS   4096
#define DD   64
#define HDIM (HH * DD)   // 1024
#define BLK  64
#define NB   (SS / BLK)  // 64
#define KSEL 7

typedef __attribute__((ext_vector_type(2))) float v2f;
typedef __attribute__((ext_vector_type(8))) float v8f;

__device__ __forceinline__ v8f wmma_f32(v2f a, v2f b, v8f c) {
  // D = A(16x4 f32) * B(4x16 f32) + C(16x16 f32); full fp32, matches reference precision.
  return __builtin_amdgcn_wmma_f32_16x16x4_f32(false, a, false, b, (short)0, c, false, false);
}

// CDNA5 async global->LDS copy (tracked by ASYNCcnt). lds_byte_off is the
// workgroup-relative LDS byte address: for a flat shared pointer the LDS
// aperture puts the offset in addr[31:0] (ISA 10.2), so truncation is exact.
__device__ __forceinline__ void async_ld_b128(unsigned lds_byte_off, const void* gptr) {
  asm volatile("global_load_async_to_lds_b128 %0, %1, off"
               :
               : "v"(lds_byte_off), "v"(gptr)
               : "memory");
}
__device__ __forceinline__ unsigned lds_off(const void* p) {
  return (unsigned)(uintptr_t)p;
}

// ---------------------------------------------------------------------------
// Kernel 1: per-block mean pooling of Q and K.  grid = B*H*NB, block = 64 (=D)
// qp/kp layout: [(b*H + h)*NB + n][d]
// ---------------------------------------------------------------------------
__global__ void sla_pool(const float* __restrict__ q, const float* __restrict__ k,
                         float* __restrict__ qp, float* __restrict__ kp) {
  int bid = blockIdx.x;            // b*H*NB + h*NB + n
  int d = threadIdx.x;             // 0..63
  int n = bid % NB;
  int h = (bid / NB) % HH;
  int b = bid / (NB * HH);

  size_t base = ((size_t)b * SS + (size_t)n * BLK) * HDIM + (size_t)h * DD + d;
  float sq = 0.f, sk = 0.f;
  for (int i = 0; i < BLK; ++i) {
    sq += q[base + (size_t)i * HDIM];
    sk += k[base + (size_t)i * HDIM];
  }
  qp[(size_t)bid * DD + d] = sq * (1.0f / (float)BLK);
  kp[(size_t)bid * DD + d] = sk * (1.0f / (float)BLK);
}

// ---------------------------------------------------------------------------
// Kernel 2: pooled block scores + per-row top-7 selection.
// grid = B*H, block = 64.  Strict '>' over repeated max passes == lax.top_k
// tie behavior (lowest index first).  Positive scale doesn't change ordering.
// ---------------------------------------------------------------------------
__global__ void sla_topk(const float* __restrict__ qp, const float* __restrict__ kp,
                         int* __restrict__ sel) {
  __shared__ float sc[NB][NB];
  int bh = blockIdx.x;
  int i = threadIdx.x;

  const float* qrow = qp + ((size_t)bh * NB + i) * DD;
  for (int j = 0; j < NB; ++j) {
    const float* krow = kp + ((size_t)bh * NB + j) * DD;
    float dot = 0.f;
#pragma unroll 8
    for (int d = 0; d < DD; ++d) dot += qrow[d] * krow[d];
    sc[i][j] = dot;
  }
  // each thread only touches its own row -> no barrier needed
  for (int t = 0; t < KSEL; ++t) {
    float best = -INFINITY;
    int bj = 0;
    for (int j = 0; j < NB; ++j) {
      float vv = sc[i][j];
      if (vv > best) { best = vv; bj = j; }
    }
    sc[i][bj] = -INFINITY;
    sel[((size_t)bh * NB + i) * KSEL + t] = bj;
  }
}

// ---------------------------------------------------------------------------
// Kernel 3: sparse attention over the 7 selected key blocks, flash-style
// online softmax, fp32 WMMA 16x16x4, double-buffered async global->LDS tiles.
// grid = B*H*NB, block = 128 (4 waves); each wave owns a 16-row query strip.
// LDS: 2x K tile + 2x V tile (64KB) + P staging (16KB) = 80KB (<320KB/WGP).
// ---------------------------------------------------------------------------
__global__ __launch_bounds__(128) void sla_attend(
    const float* __restrict__ q, const float* __restrict__ k,
    const float* __restrict__ v, const int* __restrict__ sel,
    float* __restrict__ out) {
  __shared__ __align__(16) float Ks[2][BLK][DD];
  __shared__ __align__(16) float Vs[2][BLK][DD];
  __shared__ __align__(16) float Ps[4][16][DD];

  int bid = blockIdx.x;
  int n = bid % NB;
  int h = (bid / NB) % HH;
  int b = bid / (NB * HH);

  int tid = threadIdx.x;
  int w = tid >> 5;               // wave 0..3
  int lane = tid & 31;
  int ln = lane & 15;             // M / N index within 16
  int hi = (lane >> 4) * 2;       // K offset: lanes 0-15 -> 0, lanes 16-31 -> 2
  int rowOff = (lane >> 4) * 8;   // C/D: lanes 16-31 hold rows v+8

  // Per-thread slice of the cooperative tile copy (8 x 16B for K, same for V)
  int cr = (tid * 2) >> 5;        // base row helper (recomputed per i below)

  const int* selp = sel + ((size_t)(b * HH + h) * NB + n) * KSEL;

  // Issue one tile's worth of async loads (16 instrs/thread => ASYNCcnt += 16)
  auto issue_tile = [&](int kb, int buf) {
    const float* kbase = k + ((size_t)b * SS + (size_t)kb * BLK) * HDIM + (size_t)h * DD;
    const float* vbase = v + ((size_t)b * SS + (size_t)kb * BLK) * HDIM + (size_t)h * DD;
#pragma unroll
    for (int i = 0; i < 8; ++i) {
      int e = tid + i * 128;        // float4 index 0..1023
      int r = e >> 4;               // key row 0..63
      int c = (e & 15) << 2;        // d offset 0..60 step 4
      async_ld_b128(lds_off(&Ks[buf][r][c]), kbase + (size_t)r * HDIM + c);
      async_ld_b128(lds_off(&Vs[buf][r][c]), vbase + (size_t)r * HDIM + c);
    }
  };
  (void)cr;

  // --- Preload this wave's Q strip as 16 A-fragments (v2f per lane) ---
  size_t qbase = ((size_t)b * SS + (size_t)n * BLK + (size_t)(w * 16 + ln)) * HDIM +
                 (size_t)h * DD;
  v2f qa[16];
#pragma unroll
  for (int kk = 0; kk < 16; ++kk)
    qa[kk] = *(const v2f*)(q + qbase + kk * 4 + hi);

  v8f o[4] = {};                  // O accumulator: 16 rows x 64 d, C/D layout
  float m_run[8], l_run[8];
#pragma unroll
  for (int vv = 0; vv < 8; ++vv) { m_run[vv] = -INFINITY; l_run[vv] = 0.f; }

  // Prologue: prefetch tile 0
  issue_tile(selp[0], 0);

  for (int t = 0; t < KSEL; ++t) {
    int buf = t & 1;
    if (t + 1 < KSEL) {
      // Prefetch next tile into the other buffer (its readers finished at the
      // end-of-iteration barrier of t-1), then wait for tile t only:
      // 16 outstanding async ops == the just-issued next tile.
      issue_tile(selp[t + 1], buf ^ 1);
      asm volatile("s_wait_asynccnt 0x10" ::: "memory");
    } else {
      asm volatile("s_wait_asynccnt 0x0" ::: "memory");
    }
    __syncthreads();  // all waves' portions of tile t visible in LDS

    // --- S = scale * Q K^T : 4 N-tiles of 16x16, K-loop over D=64 in steps of 4
    v8f st[4];
#pragma unroll
    for (int nt = 0; nt < 4; ++nt) {
      v8f c = {};
#pragma unroll
      for (int kk = 0; kk < 16; ++kk) {
        // B[kd][ncol] = K[ncol][kd]
        v2f bf = *(const v2f*)&Ks[buf][nt * 16 + ln][kk * 4 + hi];
        c = wmma_f32(qa[kk], bf, c);
      }
      st[nt] = c * 0.125f;            // 1/sqrt(64)
    }

    // --- online softmax update over this block's 64 columns ---
#pragma unroll
    for (int vv = 0; vv < 8; ++vv) {
      float mx = fmaxf(fmaxf(st[0][vv], st[1][vv]), fmaxf(st[2][vv], st[3][vv]));
#pragma unroll
      for (int off = 1; off < 16; off <<= 1)
        mx = fmaxf(mx, __shfl_xor(mx, off, 16));
      float mnew = fmaxf(m_run[vv], mx);
      float alpha = __expf(m_run[vv] - mnew);   // 0 on first block (m_run=-inf)
#pragma unroll
      for (int nt = 0; nt < 4; ++nt) o[nt][vv] *= alpha;
      float rs = 0.f;
#pragma unroll
      for (int nt = 0; nt < 4; ++nt) {
        float p = __expf(st[nt][vv] - mnew);
        st[nt][vv] = p;
        rs += p;
      }
#pragma unroll
      for (int off = 1; off < 16; off <<= 1)
        rs += __shfl_xor(rs, off, 16);
      l_run[vv] = l_run[vv] * alpha + rs;
      m_run[vv] = mnew;
    }

    // --- stage P (C/D layout -> row-major LDS) for A-layout reads ---
#pragma unroll
    for (int nt = 0; nt < 4; ++nt)
#pragma unroll
      for (int vv = 0; vv < 8; ++vv)
        Ps[w][vv + rowOff][nt * 16 + ln] = st[nt][vv];

    __builtin_amdgcn_wave_barrier();  // wave-private region; LDS is in-order per wave

    // --- O += P * V : A = P[16 x 64], B = V[64 x 16-per-tile] ---
#pragma unroll
    for (int nt = 0; nt < 4; ++nt) {
      v8f c = o[nt];
#pragma unroll
      for (int kk = 0; kk < 16; ++kk) {
        v2f a = *(const v2f*)&Ps[w][ln][kk * 4 + hi];
        v2f bf;
        bf.x = Vs[buf][kk * 4 + hi + 0][nt * 16 + ln];
        bf.y = Vs[buf][kk * 4 + hi + 1][nt * 16 + ln];
        c = wmma_f32(a, bf, c);
      }
      o[nt] = c;
    }
    __syncthreads();  // all waves done with buf before it is refilled at t+2
  }

  // --- normalize and write out: out[b][n*64+row][h*64+d] ---
  size_t obase = ((size_t)b * SS + (size_t)n * BLK + (size_t)(w * 16)) * HDIM +
                 (size_t)h * DD;
#pragma unroll
  for (int vv = 0; vv < 8; ++vv) {
    float inv = 1.0f / l_run[vv];
    int r = vv + rowOff;
#pragma unroll
    for (int nt = 0; nt < 4; ++nt)
      out[obase + (size_t)r * HDIM + nt * 16 + ln] = o[nt][vv] * inv;
  }
}

// ---------------------------------------------------------------------------
extern "C" void kernel_launch(void* const* d_in, const int* in_sizes, int n_in,
                              void* d_out, int out_size, void* d_ws, size_t ws_size,
                              hipStream_t stream) {
  (void)in_sizes; (void)n_in; (void)out_size; (void)ws_size;
  const float* q = (const float*)d_in[0];
  const float* k = (const float*)d_in[1];
  const float* v = (const float*)d_in[2];
  float* out = (float*)d_out;

  // workspace: q_pool (512KB) | k_pool (512KB) | sel indices (56KB)
  float* qp = (float*)d_ws;
  float* kp = qp + (size_t)BB * HH * NB * DD;
  int* sel = (int*)(kp + (size_t)BB * HH * NB * DD);

  sla_pool<<<BB * HH * NB, DD, 0, stream>>>(q, k, qp, kp);
  sla_topk<<<BB * HH, NB, 0, stream>>>(qp, kp, sel);
  sla_attend<<<BB * HH * NB, 128, 0, stream>>>(q, k, v, sel, out);
}